// PyQMatrixBlock_12678743458164
// MI455X (gfx1250) — compile-verified
//
#include <hip/hip_runtime.h>
#include <stdint.h>

// Quantum 2-qubit gate application, N_QUBITS=24, QUBITS=(5,12), BATCH=4.
// out[i, r] = sum_j M[i,j] * s[j, r], j axis at flat-index bits 20 (q5) and 13 (q12).
// Memory-bound: 512 MiB traffic -> ~23us floor at 23.3 TB/s. WMMA f32 16x16x4
// does the whole 4x4 contraction for 16 columns per instruction.

typedef __attribute__((ext_vector_type(2))) float v2f;
typedef __attribute__((ext_vector_type(8))) float v8f;

#define STRIDE_J2 8192u          // qubit 12 -> flat bit 13
#define STRIDE_J1 (1u << 20)     // qubit 5  -> flat bit 20
#define NCOLS     (1u << 24)     // 2^22 rest * 4 batch
#define NTILES    (NCOLS / 16u)  // 16 columns per WMMA tile
#define TILES_PER_ITER 4

// Re-insert the two "hole" bits (13 and 20) into a packed 24-bit column index.
__device__ __forceinline__ uint32_t expand_r(uint32_t r) {
    uint32_t t = ((r & 0xFFFFE000u) << 1) | (r & 0x1FFFu);       // hole at bit 13
    return ((t & 0xFFF00000u) << 1) | (t & 0x000FFFFFu);         // hole at bit 20
}

__global__ __launch_bounds__(256) void gate_wmma_kernel(
    const float* __restrict__ state,   // 2^26 floats
    const float* __restrict__ mat,     // 16 floats, M[i][j] = mat[i*4+j]
    float* __restrict__ out)           // 2^26 floats
{
    const uint32_t lane   = threadIdx.x & 31u;
    const uint32_t wave   = blockIdx.x * (blockDim.x >> 5) + (threadIdx.x >> 5);
    const uint32_t nwaves = gridDim.x * (blockDim.x >> 5);

    // A-matrix (16x4 f32, 2 VGPRs): gate rows replicated 4x down M.
    // Lane L<16: A[L, K=0/1]; lane L>=16: A[L-16, K=2/3].
    const uint32_t row     = lane & 3u;
    const uint32_t colbase = (lane >> 4) << 1;
    v2f A;
    A.x = mat[row * 4u + colbase];
    A.y = mat[row * 4u + colbase + 1u];

    const uint32_t n     = lane & 15u;   // column within tile
    const uint32_t khalf = lane >> 4;    // 0 -> K=0/1, 1 -> K=2/3

    for (uint32_t t0 = wave * TILES_PER_ITER; t0 < NTILES;
         t0 += nwaves * TILES_PER_ITER) {
        uint32_t base[TILES_PER_ITER];
        v2f      B[TILES_PER_ITER];

        // Issue all loads first (MLP). B.x = s[K=2*khalf, r], B.y = s[K=2*khalf+1, r].
        #pragma unroll
        for (int u = 0; u < TILES_PER_ITER; ++u) {
            uint32_t r = (t0 + (uint32_t)u) * 16u + n;
            uint32_t f = expand_r(r) + khalf * STRIDE_J1;
            base[u] = f;
            B[u].x = __builtin_nontemporal_load(state + f);
            B[u].y = __builtin_nontemporal_load(state + f + STRIDE_J2);
        }

        v8f D[TILES_PER_ITER];
        #pragma unroll
        for (int u = 0; u < TILES_PER_ITER; ++u) {
            v8f c = {};
            // (neg_a, A, neg_b, B, c_mod, C, reuse_a, reuse_b)
            D[u] = __builtin_amdgcn_wmma_f32_16x16x4_f32(
                false, A, false, B[u], (short)0, c, false, false);
        }

        // D VGPRs 0..3 on lanes 0..15 hold the 4 unique output rows i=0..3
        // (M=4..15 are the redundant gate replicas). khalf==0 => base==expand(r).
        if (lane < 16u) {
            #pragma unroll
            for (int u = 0; u < TILES_PER_ITER; ++u) {
                uint32_t f = base[u];
                __builtin_nontemporal_store(D[u][0], out + f);                          // i=0
                __builtin_nontemporal_store(D[u][1], out + f + STRIDE_J2);              // i=1
                __builtin_nontemporal_store(D[u][2], out + f + STRIDE_J1);              // i=2
                __builtin_nontemporal_store(D[u][3], out + f + STRIDE_J1 + STRIDE_J2);  // i=3
            }
        }
    }
}

extern "C" void kernel_launch(void* const* d_in, const int* in_sizes, int n_in,
                              void* d_out, int out_size, void* d_ws, size_t ws_size,
                              hipStream_t stream) {
    (void)in_sizes; (void)n_in; (void)out_size; (void)d_ws; (void)ws_size;
    const float* state = (const float*)d_in[0];  // 2^26 floats
    const float* mat   = (const float*)d_in[1];  // (4,4,1) -> 16 floats
    float*       out   = (float*)d_out;          // 2^26 floats

    const int block = 256;   // 8 wave32s
    const int grid  = 4096;  // 32768 waves; 2^18 tile-chunks -> 8 chunks/wave
    gate_wmma_kernel<<<grid, block, 0, stream>>>(state, mat, out);
}